// CausalSelfAttention_TSSA_63977832841579
// MI455X (gfx1250) — compile-verified
//
#include <hip/hip_runtime.h>
#include <hip/hip_bf16.h>

// ---------------------------------------------------------------------------
// Problem constants
// ---------------------------------------------------------------------------
#define BB    4
#define NN    2048
#define CC    1024
#define HH    16
#define DD    64
#define MROWS (BB * NN)          // 8192
#define KDIM  CC                 // 1024
#define BLOCK_SEQ 2048
#define EPSF  1.1920929e-07f

// sequence chunking for the blocked scans
#define CH    16                 // chunks per sequence
#define CL    (NN / CH)          // 128 n per chunk

// GEMM K-chunking / LDS layout (TDM-staged)
#define KC      32               // K elements per chunk
#define KCH     (KDIM / KC)      // 32 chunks
#define LDS_ROW (KC + 2)         // 34 dwords: TDM pad (interval 32dw, amount 2dw)
#define ASZ     (16  * LDS_ROW)  // A tile floats (544)
#define BSZ     (128 * LDS_ROW)  // B tile floats (4352)
#define BUFSZ   (ASZ + BSZ)      // 4896 floats per buffer

typedef __attribute__((ext_vector_type(2))) float v2f;
typedef __attribute__((ext_vector_type(8))) float v8f;
typedef __attribute__((ext_vector_type(4))) unsigned int u32x4;
typedef __attribute__((ext_vector_type(8))) int i32x8;
typedef __attribute__((ext_vector_type(4))) int i32x4;

// ---------------------------------------------------------------------------
// TDM: async-load a 2D tile (tile_rows x tile_k fp32, row-major, k contiguous)
// from global into LDS, appending 2 pad DWORDs after every 32 data DWORDs
// (-> LDS row stride = 34 DWORDs, bank-conflict-free, 8B-aligned frags).
// ---------------------------------------------------------------------------
__device__ __forceinline__ void tdm_load_tile_2d(unsigned lds_byte_off,
                                                 const float* gptr,
                                                 unsigned tile_k, unsigned tile_rows,
                                                 unsigned tensor_k, unsigned tensor_rows,
                                                 unsigned row_stride_elems)
{
    unsigned long long ga = (unsigned long long)(uintptr_t)gptr;
    u32x4 g0;
    g0[0] = 1u;                                               // count=1 (valid user D#)
    g0[1] = lds_byte_off;                                     // lds_addr
    g0[2] = (unsigned)(ga & 0xFFFFFFFFull);                   // global_addr[31:0]
    g0[3] = (unsigned)((ga >> 32) & 0x01FFFFFFull)            // global_addr[56:32]
          | (2u << 30);                                       // type=2 ("image")
    i32x8 g1;
    g1[0] = (int)((2u << 16)      // data_size = 4 bytes
                | (1u << 20)      // pad_enable
                | (4u << 22)      // pad_interval: 32 DWORDs
                | (1u << 25));    // pad_amount : 2 DWORDs
    g1[1] = (int)((tensor_k & 0xFFFFu) << 16);                            // dim0 lo16
    g1[2] = (int)(((tensor_k >> 16) & 0xFFFFu) | ((tensor_rows & 0xFFFFu) << 16));
    g1[3] = (int)(((tensor_rows >> 16) & 0xFFFFu) | ((tile_k & 0xFFFFu) << 16));
    g1[4] = (int)(tile_rows & 0xFFFFu);                       // tile_dim1 (tile_dim2=0)
    g1[5] = (int)row_stride_elems;                            // tensor_dim0_stride lo32
    g1[6] = 0;
    g1[7] = 0;
    i32x4 gz4 = {0, 0, 0, 0};
    i32x8 gz8 = {0, 0, 0, 0, 0, 0, 0, 0};
    __builtin_amdgcn_tensor_load_to_lds(g0, g1, gz4, gz4, gz8, 0);
}

// ---------------------------------------------------------------------------
// GEMM: C[m,o] = sum_k A[m,k] * W[o,k]  (fp32 WMMA, TDM double-buffered LDS)
//   Block = 256 thr (8 waves) -> 16(M) x 128(N) strip; wave w owns cols
//   [blockIdx.y*128 + 16w, +16). grid = (MROWS/16, CC/128)
// ---------------------------------------------------------------------------
__global__ __launch_bounds__(256)
void gemm_nt_wmma_tdm(const float* __restrict__ A,
                      const float* __restrict__ W,
                      float* __restrict__ Cout)
{
    __shared__ __align__(16) float smem[2 * BUFSZ];

    const int lane = threadIdx.x & 31;
    const int wave = threadIdx.x >> 5;
    const int l15  = lane & 15;
    const int half = lane >> 4;

    const int rowBase  = blockIdx.x * 16;
    const int colBlock = blockIdx.y * 128;

    const unsigned smemBase = (unsigned)(uintptr_t)(void*)smem;
    const bool issuer = (wave == 0);

    // prologue: stage k-chunk 0
    if (issuer) {
        tdm_load_tile_2d(smemBase,            A + (size_t)rowBase  * KDIM,
                         KC, 16,  KDIM, MROWS, KDIM);
        tdm_load_tile_2d(smemBase + ASZ * 4,  W + (size_t)colBlock * KDIM,
                         KC, 128, KDIM, CC,    KDIM);
        __builtin_amdgcn_s_wait_tensorcnt(0);
    }
    __syncthreads();

    v8f acc = {};

    for (int chunk = 0; chunk < KCH; ++chunk) {
        const int cur = chunk & 1;

        // issue async TDM loads for the next k-chunk into the other buffer
        if (chunk + 1 < KCH && issuer) {
            const unsigned nb = smemBase + (unsigned)((cur ^ 1) * BUFSZ * 4);
            const size_t ko = (size_t)(chunk + 1) * KC;
            tdm_load_tile_2d(nb,           A + (size_t)rowBase  * KDIM + ko,
                             KC, 16,  KDIM, MROWS, KDIM);
            tdm_load_tile_2d(nb + ASZ * 4, W + (size_t)colBlock * KDIM + ko,
                             KC, 128, KDIM, CC,    KDIM);
        }

        const float* bufA = smem + cur * BUFSZ;
        const float* bufB = bufA + ASZ;
        const float* aRow = bufA + l15 * LDS_ROW + half * 2;
        const float* bRow = bufB + (wave * 16 + l15) * LDS_ROW + half * 2;

#pragma unroll
        for (int kk = 0; kk < KC; kk += 4) {
            v2f a = *(const v2f*)(aRow + kk);
            v2f b = *(const v2f*)(bRow + kk);
            acc = __builtin_amdgcn_wmma_f32_16x16x4_f32(
                false, a, false, b, (short)0, acc, false, false);
        }

        if (chunk + 1 < KCH) {
            if (issuer) __builtin_amdgcn_s_wait_tensorcnt(0);
            __syncthreads();
        }
    }

    // C/D layout: VGPR r -> M = half*8 + r, N = l15
    float* cRow = Cout + (size_t)(rowBase + half * 8) * CC + colBlock + wave * 16 + l15;
#pragma unroll
    for (int r = 0; r < 8; ++r)
        cRow[(size_t)r * CC] = acc[r];
}

// ---------------------------------------------------------------------------
// Scan A, pass 1: per-channel per-chunk sums of w^2.
//   grid = (B*H, CH), block = 64 (thread = d). chunkA[(bh*DD+d)*CH + c]
// ---------------------------------------------------------------------------
__global__ __launch_bounds__(64)
void chunk_sq_sums(const float* __restrict__ w1, float* __restrict__ chunkA)
{
    const int bh = blockIdx.x, c = blockIdx.y, d = threadIdx.x;
    const int b = bh / HH, h = bh % HH;
    const float* p = w1 + ((size_t)b * NN + c * CL) * CC + h * DD + d;
    float s = 0.0f;
    for (int i = 0; i < CL; ++i) {
        float v = p[(size_t)i * CC];
        s += v * v;
    }
    chunkA[((size_t)bh * DD + d) * CH + c] = s;
}

// ---------------------------------------------------------------------------
// Scan A, pass 2: within-chunk causal scan + head statistic.
//   grid = (B*H, CH), block = 32. Lane owns d and d+32.
// ---------------------------------------------------------------------------
__global__ __launch_bounds__(32)
void phase_a_scan(const float* __restrict__ w1,
                  const float* __restrict__ chunkA,
                  const float* __restrict__ temp,
                  const float* __restrict__ dbias,
                  float* __restrict__ tmpOut)          // (B,H,N)
{
    const int bh = blockIdx.x, c = blockIdx.y, lane = threadIdx.x;
    const int b = bh / HH, h = bh % HH;

    float cum0 = 0.0f, cum1 = 0.0f;
    {
        const float* s0 = chunkA + ((size_t)bh * DD + lane) * CH;
        const float* s1 = chunkA + ((size_t)bh * DD + lane + 32) * CH;
        for (int j = 0; j < c; ++j) { cum0 += s0[j]; cum1 += s1[j]; }
    }

    const float th = temp[h];
    const float* db = dbias + (size_t)h * BLOCK_SEQ;
    const float* base = w1 + (size_t)b * NN * CC + h * DD;
    float* out = tmpOut + (size_t)bh * NN;

    for (int n = c * CL; n < (c + 1) * CL; ++n) {
        const float* p = base + (size_t)n * CC;
        float w0 = p[lane], w1v = p[lane + 32];
        float s0 = w0 * w0, s1 = w1v * w1v;
        cum0 += s0; cum1 += s1;
        float contrib = s0 / fmaxf(cum0, EPSF) + s1 / fmaxf(cum1, EPSF);
        for (int off = 16; off > 0; off >>= 1)
            contrib += __shfl_xor(contrib, off, 32);
        if (lane == 0)
            out[n] = (contrib + (float)DD * db[n]) * th;
    }
}

// ---------------------------------------------------------------------------
// Softmax over heads (H=16) of tmp (B,H,N), in place -> Pi.
// ---------------------------------------------------------------------------
__global__ __launch_bounds__(256)
void softmax_heads(float* __restrict__ t)
{
    int idx = blockIdx.x * blockDim.x + threadIdx.x;   // b*N + n
    if (idx >= BB * NN) return;
    int b = idx / NN, n = idx % NN;
    float* p = t + (size_t)b * HH * NN + n;

    float mx = -3.402823466e+38f;
#pragma unroll
    for (int h = 0; h < HH; ++h) mx = fmaxf(mx, p[(size_t)h * NN]);
    float sum = 0.0f, e[HH];
#pragma unroll
    for (int h = 0; h < HH; ++h) { e[h] = __expf(p[(size_t)h * NN] - mx); sum += e[h]; }
    float inv = 1.0f / sum;
#pragma unroll
    for (int h = 0; h < HH; ++h) p[(size_t)h * NN] = e[h] * inv;
}

// ---------------------------------------------------------------------------
// Pi chunk sums: piCS[bh*CH + c] = sum over chunk of Pi. 1024 work items.
// ---------------------------------------------------------------------------
__global__ __launch_bounds__(256)
void pi_chunk_sums(const float* __restrict__ Pi, float* __restrict__ piCS)
{
    int t = blockIdx.x * blockDim.x + threadIdx.x;
    if (t >= BB * HH * CH) return;
    int bh = t / CH, c = t % CH;
    const float* p = Pi + (size_t)bh * NN + c * CL;
    float s = 0.0f;
    for (int i = 0; i < CL; ++i) s += p[i];
    piCS[t] = s;
}

// ---------------------------------------------------------------------------
// Scan B, pass 1: per-channel per-chunk sums of w^2 * Pi.
// ---------------------------------------------------------------------------
__global__ __launch_bounds__(64)
void chunk_wsqpi_sums(const float* __restrict__ w1,
                      const float* __restrict__ Pi,
                      float* __restrict__ chunkB)
{
    const int bh = blockIdx.x, c = blockIdx.y, d = threadIdx.x;
    const int b = bh / HH, h = bh % HH;
    const float* p  = w1 + ((size_t)b * NN + c * CL) * CC + h * DD + d;
    const float* pp = Pi + (size_t)bh * NN + c * CL;
    float s = 0.0f;
    for (int i = 0; i < CL; ++i) {
        float v = p[(size_t)i * CC];
        s += v * v * pp[i];
    }
    chunkB[((size_t)bh * DD + d) * CH + c] = s;
}

// ---------------------------------------------------------------------------
// Scan B, pass 2: y = -(w*Pi)/(1 + cum(w^2*Pi)/(cum(Pi)+eps)), in place on w1.
//   grid = (B*H, CH), block = 32.
// ---------------------------------------------------------------------------
__global__ __launch_bounds__(32)
void phase_b_scan(float* __restrict__ w1,
                  const float* __restrict__ Pi,
                  const float* __restrict__ chunkB,
                  const float* __restrict__ piCS)
{
    const int bh = blockIdx.x, c = blockIdx.y, lane = threadIdx.x;
    const int b = bh / HH, h = bh % HH;

    float cw0 = 0.0f, cw1 = 0.0f, cumPi = 0.0f;
    {
        const float* s0 = chunkB + ((size_t)bh * DD + lane) * CH;
        const float* s1 = chunkB + ((size_t)bh * DD + lane + 32) * CH;
        const float* ps = piCS + (size_t)bh * CH;
        for (int j = 0; j < c; ++j) { cw0 += s0[j]; cw1 += s1[j]; cumPi += ps[j]; }
    }

    float* base = w1 + (size_t)b * NN * CC + h * DD;
    const float* pi = Pi + (size_t)bh * NN;

    for (int n = c * CL; n < (c + 1) * CL; ++n) {
        float p = pi[n];
        cumPi += p;
        float invDen = 1.0f / (cumPi + EPSF);

        float* q = base + (size_t)n * CC;
        float w0 = q[lane], w1v = q[lane + 32];
        cw0 += w0 * w0 * p;
        cw1 += w1v * w1v * p;
        q[lane]      = -(w0  * p) / (1.0f + cw0 * invDen);
        q[lane + 32] = -(w1v * p) / (1.0f + cw1 * invDen);
    }
}

// ---------------------------------------------------------------------------
// Host-side launcher
// ---------------------------------------------------------------------------
extern "C" void kernel_launch(void* const* d_in, const int* in_sizes, int n_in,
                              void* d_out, int out_size, void* d_ws, size_t ws_size,
                              hipStream_t stream)
{
    const float* x     = (const float*)d_in[0];
    const float* Wattn = (const float*)d_in[1];
    const float* Wproj = (const float*)d_in[2];
    const float* temp  = (const float*)d_in[3];
    const float* dbias = (const float*)d_in[4];
    float* out         = (float*)d_out;

    float* w1     = (float*)d_ws;                         // 32 MB  (B,N,C)
    float* tmp    = w1   + (size_t)MROWS * CC;            // 512 KB (B,H,N): tmp/Pi
    float* chunkA = tmp  + (size_t)BB * HH * NN;          // 256 KB
    float* chunkB = chunkA + (size_t)BB * HH * DD * CH;   // 256 KB
    float* piCS   = chunkB + (size_t)BB * HH * DD * CH;   // 4 KB

    dim3 ggrid(MROWS / 16, CC / 128);
    dim3 sgrid(BB * HH, CH);

    // 1) w = x @ W_attn^T  (WMMA + TDM staging)
    gemm_nt_wmma_tdm<<<ggrid, 256, 0, stream>>>(x, Wattn, w1);
    // 2) blocked causal scan of w^2 -> tmp
    chunk_sq_sums<<<sgrid, 64, 0, stream>>>(w1, chunkA);
    phase_a_scan<<<sgrid, 32, 0, stream>>>(w1, chunkA, temp, dbias, tmp);
    // 3) softmax over heads -> Pi (in place)
    softmax_heads<<<(BB * NN + 255) / 256, 256, 0, stream>>>(tmp);
    // 4) blocked causal scan with Pi -> y (in place on w1)
    pi_chunk_sums<<<(BB * HH * CH + 255) / 256, 256, 0, stream>>>(tmp, piCS);
    chunk_wsqpi_sums<<<sgrid, 64, 0, stream>>>(w1, tmp, chunkB);
    phase_b_scan<<<sgrid, 32, 0, stream>>>(w1, tmp, chunkB, piCS);
    // 5) out = y @ W_proj^T
    gemm_nt_wmma_tdm<<<ggrid, 256, 0, stream>>>(w1, Wproj, out);
}